// GromovWassersteinDistance_31353261261216
// MI455X (gfx1250) — compile-verified
//
#include <hip/hip_runtime.h>
#include <math.h>

// ---------------- problem constants ----------------
#define B_   2
#define M_   768
#define N_   512
#define MN_  ((long)M_ * (long)N_)
#define EPS_ 0.05f
#define OUTER_ 10
#define INNER_ 20

typedef float v2f __attribute__((ext_vector_type(2)));
typedef float v8f __attribute__((ext_vector_type(8)));

#ifndef __has_builtin
#define __has_builtin(x) 0
#endif

#if defined(__HIP_DEVICE_COMPILE__) && __has_builtin(__builtin_amdgcn_global_load_async_to_lds_b128)
#define USE_ASYNC_LDS 1
#else
#define USE_ASYNC_LDS 0
#endif

#if USE_ASYNC_LDS
// Builtin signature (from hipcc diagnostic): arg0 is `int __vector(4)` pointer in
// the global (__device__, AS1) address space; arg1 is the LDS-side pointer (AS3).
typedef int v4i_ __attribute__((vector_size(16)));
typedef __attribute__((address_space(1))) v4i_* g_v4i_ptr;
typedef __attribute__((address_space(3))) v4i_* l_v4i_ptr;
#endif

// 16-byte global -> LDS copy (async-to-LDS on gfx1250 when available)
__device__ __forceinline__ void copy16(const float* __restrict__ g, float* l) {
#if USE_ASYNC_LDS
  v4i_* gg = (v4i_*)const_cast<float*>(g);  // reinterpret pointee (generic AS)
  v4i_* ll = (v4i_*)l;
  __builtin_amdgcn_global_load_async_to_lds_b128(
      (g_v4i_ptr)gg, (l_v4i_ptr)ll, 0, 0);
#else
  float4 tmp = *(const float4*)g;
  *(float4*)l = tmp;
#endif
}

__device__ __forceinline__ void wait_async() {
#if USE_ASYNC_LDS
#if __has_builtin(__builtin_amdgcn_s_wait_asynccnt)
  __builtin_amdgcn_s_wait_asynccnt(0);
#else
  asm volatile("s_wait_asynccnt 0" ::: "memory");
#endif
#endif
}

// ---------------- fp32 WMMA GEMM ----------------
// C[MdimxNdim] = A[MdimxKdim] * B[KdimxNdim], row-major, batched over blockIdx.z.
// mode 0: C = acc
// mode 1: C = (2*acc - t1[row] - t2[col]) * scale     (fused logK = -G/eps epilogue)
#define BM 64
#define BN 64
#define BK 16
#define ASTR 20   // padded LDS row stride for A (16B aligned, conflict-free)
#define BSTR 80   // padded LDS row stride for B
#define GEMM_THREADS 128

__global__ __launch_bounds__(GEMM_THREADS)
void k_gemm(const float* __restrict__ Ag, const float* __restrict__ Bg,
            float* __restrict__ Cg,
            int Mdim, int Ndim, int Kdim,
            long strideA, long strideB, long strideC,
            int lda, int ldb, int ldc,
            int mode, const float* __restrict__ t1v, const float* __restrict__ t2v,
            float scale)
{
  __shared__ float Al[2][BM * ASTR];
  __shared__ float Bl[2][BK * BSTR];

  const int bz = blockIdx.z;
  const int m0 = blockIdx.y * BM;
  const int n0 = blockIdx.x * BN;
  const float* A  = Ag + (long)bz * strideA;
  const float* Bm = Bg + (long)bz * strideB;
  float*       C  = Cg + (long)bz * strideC;

  const int t    = threadIdx.x;
  const int lane = t & 31;
  const int wv   = t >> 5;
  const int hlf  = lane >> 4;     // 0/1: which half-wave (selects K pair / M half)
  const int lm   = lane & 15;
  const int wrow = (wv >> 1) * 32;
  const int wcol = (wv & 1) * 32;

  const v8f vzero = {0.f, 0.f, 0.f, 0.f, 0.f, 0.f, 0.f, 0.f};
  v8f acc[2][2];
  acc[0][0] = vzero; acc[0][1] = vzero; acc[1][0] = vzero; acc[1][1] = vzero;

  const int nk = Kdim / BK;

  auto load_chunk = [&](int kc, int buf) {
    const int kb = kc * BK;
    // A tile: 64x16 floats = 256 float4, 2 per thread
#pragma unroll
    for (int vv = 0; vv < 2; ++vv) {
      const int id = t + vv * GEMM_THREADS;   // 0..255
      const int ar = id >> 2;                 // 0..63
      const int ak = (id & 3) * 4;            // 0,4,8,12
      copy16(A + (long)(m0 + ar) * lda + (kb + ak), &Al[buf][ar * ASTR + ak]);
    }
    // B tile: 16x64 floats = 256 float4, 2 per thread
#pragma unroll
    for (int vv = 0; vv < 2; ++vv) {
      const int id = t + vv * GEMM_THREADS;
      const int br = id >> 4;                 // 0..15
      const int bn = (id & 15) * 4;           // 0..60
      copy16(Bm + (long)(kb + br) * ldb + (n0 + bn), &Bl[buf][br * BSTR + bn]);
    }
  };

  load_chunk(0, 0);
  wait_async();
  __syncthreads();

  for (int kc = 0; kc < nk; ++kc) {
    const int buf = kc & 1;
    if (kc + 1 < nk) load_chunk(kc + 1, buf ^ 1);  // overlap next-chunk DMA with math

#pragma unroll
    for (int k0 = 0; k0 < BK; k0 += 4) {
      v2f fa[2], fb[2];
      // A frag (16x4 fp32): lanes 0-15 -> K = k0,k0+1 ; lanes 16-31 -> K = k0+2,k0+3
#pragma unroll
      for (int i = 0; i < 2; ++i) {
        const float* p = &Al[buf][(wrow + i * 16 + lm) * ASTR + k0 + hlf * 2];
        fa[i][0] = p[0]; fa[i][1] = p[1];
      }
      // B frag (4x16 fp32): row striped across lanes within a VGPR
#pragma unroll
      for (int j = 0; j < 2; ++j) {
        const float* p = &Bl[buf][(k0 + hlf * 2) * BSTR + wcol + j * 16 + lm];
        fb[j][0] = p[0]; fb[j][1] = p[BSTR];
      }
#pragma unroll
      for (int i = 0; i < 2; ++i)
#pragma unroll
        for (int j = 0; j < 2; ++j)
          acc[i][j] = __builtin_amdgcn_wmma_f32_16x16x4_f32(
              false, fa[i], false, fb[j], (short)0, acc[i][j], false, false);
    }
    wait_async();
    __syncthreads();
  }

  // epilogue: C/D layout -> VGPR r holds M = r + 8*hlf, N = lm
#pragma unroll
  for (int i = 0; i < 2; ++i) {
#pragma unroll
    for (int j = 0; j < 2; ++j) {
      const int r0 = m0 + wrow + i * 16 + hlf * 8;
      const int c  = n0 + wcol + j * 16 + lm;
      const float t2c = (mode == 1) ? t2v[(long)bz * Ndim + c] : 0.f;
#pragma unroll
      for (int r = 0; r < 8; ++r) {
        float val = acc[i][j][r];
        const int row = r0 + r;
        if (mode == 1)
          val = (2.f * val - t1v[(long)bz * Mdim + row] - t2c) * scale;
        C[(long)row * ldc + c] = val;
      }
    }
  }
}

// ---------------- reduction helpers ----------------
__device__ __forceinline__ float wave_sum(float v) {
#pragma unroll
  for (int off = 16; off; off >>= 1) v += __shfl_xor(v, off, 32);
  return v;
}

template <int BS>
__device__ __forceinline__ float block_sum(float val) {
  __shared__ float sm[BS / 32];
  val = wave_sum(val);
  const int wv = threadIdx.x >> 5, lane = threadIdx.x & 31;
  if (lane == 0) sm[wv] = val;
  __syncthreads();
  float r = (threadIdx.x < BS / 32) ? sm[threadIdx.x] : 0.f;
  if (wv == 0) r = wave_sum(r);
  return r;  // valid in thread 0
}

// rowm[b,i] = sum_j P[b,i,j]  (one wave32 per row)
__global__ __launch_bounds__(256)
void k_row_sum(const float* __restrict__ P, float* __restrict__ rowm) {
  const int lane = threadIdx.x & 31;
  const int row  = blockIdx.x * 8 + (threadIdx.x >> 5);  // 0..B*M-1
  const float* p = P + (long)row * N_;
  float s = 0.f;
  for (int j = lane; j < N_; j += 32) s += p[j];
  s = wave_sum(s);
  if (lane == 0) rowm[row] = s;
}

// colm[b,j] = sum_i P[b,i,j]  (coalesced column streaming)
__global__ __launch_bounds__(256)
void k_col_sum(const float* __restrict__ P, float* __restrict__ colm) {
  const int j = blockIdx.x * blockDim.x + threadIdx.x;
  const int b = blockIdx.y;
  const float* p = P + (long)b * MN_;
  float s = 0.f;
  for (int i = 0; i < M_; ++i) s += p[(long)i * N_ + j];
  colm[b * N_ + j] = s;
}

// y[b,r] = sum_k D[b,r,k]^2 * x[b,k]   (square on the fly; one wave per row)
__global__ __launch_bounds__(256)
void k_matvec_sq(const float* __restrict__ D, const float* __restrict__ x,
                 float* __restrict__ y, int R, int L) {
  const int lane = threadIdx.x & 31;
  const int row  = blockIdx.x * 8 + (threadIdx.x >> 5);  // 0..B*R-1
  const int b = row / R, r = row % R;
  const float* dr = D + ((long)b * R + r) * L;
  const float* xb = x + (long)b * L;
  float s = 0.f;
  for (int k = lane; k < L; k += 32) { float d = dr[k]; s += d * d * xb[k]; }
  s = wave_sum(s);
  if (lane == 0) y[row] = s;
}

// u[b,i] = log_a - LSE_j(logK[b,i,j] + v[b,j])   (online LSE, wave per row)
__global__ __launch_bounds__(256)
void k_sink_row(const float* __restrict__ logK, const float* __restrict__ v,
                float* __restrict__ u, float log_a) {
  const int lane = threadIdx.x & 31;
  const int row  = blockIdx.x * 8 + (threadIdx.x >> 5);
  const int b = row / M_;
  const float* kr = logK + (long)row * N_;
  const float* vb = v + b * N_;
  float m = -3.0e38f, s = 0.f;
  for (int j = lane; j < N_; j += 32) {
    float x = kr[j] + vb[j];
    if (x > m) { s = s * __expf(m - x) + 1.f; m = x; }
    else         s += __expf(x - m);
  }
#pragma unroll
  for (int off = 16; off; off >>= 1) {
    float m2 = __shfl_xor(m, off, 32);
    float s2 = __shfl_xor(s, off, 32);
    float mx = fmaxf(m, m2);
    s = s * __expf(m - mx) + s2 * __expf(m2 - mx);
    m = mx;
  }
  if (lane == 0) u[row] = log_a - (m + __logf(s));
}

// v[b,j] = log_b - LSE_i(logK[b,i,j] + u[b,i])   (thread per column, coalesced)
__global__ __launch_bounds__(256)
void k_sink_col(const float* __restrict__ logK, const float* __restrict__ u,
                float* __restrict__ v, float log_b) {
  const int j = blockIdx.x * blockDim.x + threadIdx.x;
  const int b = blockIdx.y;
  const float* kb = logK + (long)b * MN_;
  const float* ub = u + b * M_;
  float m = -3.0e38f, s = 0.f;
  for (int i = 0; i < M_; ++i) {
    float x = kb[(long)i * N_ + j] + ub[i];
    if (x > m) { s = s * __expf(m - x) + 1.f; m = x; }
    else         s += __expf(x - m);
  }
  v[b * N_ + j] = log_b - (m + __logf(s));
}

__global__ void k_fill(float* p, float val, long n) {
  long i = (long)blockIdx.x * blockDim.x + threadIdx.x;
  if (i < n) p[i] = val;
}

// P = exp(u + logK + v)
__global__ void k_expP(const float* __restrict__ logK, const float* __restrict__ u,
                       const float* __restrict__ v, float* __restrict__ P) {
  long idx = (long)blockIdx.x * blockDim.x + threadIdx.x;
  if (idx >= (long)B_ * MN_) return;
  long b = idx / MN_;
  long rem = idx - b * MN_;
  long i = rem / N_, j = rem - i * N_;
  P[idx] = __expf(u[b * M_ + i] + logK[idx] + v[b * N_ + j]);
}

__global__ void k_copy(const float* __restrict__ src, float* __restrict__ dst, long n) {
  long i = (long)blockIdx.x * blockDim.x + threadIdx.x;
  if (i < n) dst[i] = src[i];
}

__global__ __launch_bounds__(256)
void k_dot(const float* __restrict__ x, const float* __restrict__ y,
           float* __restrict__ out, int L) {
  const int b = blockIdx.x;
  float s = 0.f;
  for (int k = threadIdx.x; k < L; k += 256) s += x[b * L + k] * y[b * L + k];
  s = block_sum<256>(s);
  if (threadIdx.x == 0) out[b] = s;
}

__global__ __launch_bounds__(256)
void k_cross(const float* __restrict__ T, const float* __restrict__ U,
             float* __restrict__ out) {
  const int b = blockIdx.x;
  const float* tb = T + (long)b * MN_;
  const float* ub = U + (long)b * MN_;
  float s = 0.f;
  for (long k = threadIdx.x; k < MN_; k += 256) s += tb[k] * ub[k];
  s = block_sum<256>(s);
  if (threadIdx.x == 0) out[b] = s;
}

__global__ void k_final(const float* __restrict__ sc, float* __restrict__ out) {
  int b = threadIdx.x;
  if (b < B_) out[b] = sc[b] + sc[2 + b] - 2.f * sc[4 + b];
}

// ---------------- host orchestration ----------------
extern "C" void kernel_launch(void* const* d_in, const int* in_sizes, int n_in,
                              void* d_out, int out_size, void* d_ws, size_t ws_size,
                              hipStream_t stream) {
  (void)in_sizes; (void)n_in; (void)out_size; (void)ws_size;
  const float* D1 = (const float*)d_in[0];  // (B, M, M)
  const float* D2 = (const float*)d_in[1];  // (B, N, N)
  float* out = (float*)d_out;               // [dist(B), P(B*M*N)]
  float* ws  = (float*)d_ws;

  float* P  = ws;                 // B*MN
  float* T  = P  + B_ * MN_;      // B*MN : D1@P
  float* LK = T  + B_ * MN_;      // B*MN : logK, reused as U=P@D2 at the end
  float* u  = LK + B_ * MN_;      // B*M
  float* v  = u  + B_ * M_;       // B*N
  float* rm = v  + B_ * N_;       // B*M
  float* cm = rm + B_ * M_;       // B*N
  float* t1 = cm + B_ * N_;       // B*M
  float* t2 = t1 + B_ * M_;       // B*N
  float* sc = t2 + B_ * N_;       // 8 scalars: dot1[2], dot2[2], cross[2]

  const float log_a  = -logf((float)M_);
  const float log_b  = -logf((float)N_);
  const float invEps = 1.0f / EPS_;
  const long  totP   = (long)B_ * MN_;

  dim3 bl256(256);
  dim3 gemmGrid(N_ / BN, M_ / BM, B_);
  dim3 gemmBlock(GEMM_THREADS);
  dim3 rowGridM(B_ * M_ / 8), rowGridN(B_ * N_ / 8);
  dim3 colGrid(N_ / 256, B_);
  dim3 ewGrid((unsigned)((totP + 255) / 256));

  // P0 = independent coupling
  k_fill<<<ewGrid, bl256, 0, stream>>>(P, 1.0f / ((float)M_ * (float)N_), totP);

  for (int it = 0; it < OUTER_; ++it) {
    k_row_sum<<<rowGridM, bl256, 0, stream>>>(P, rm);
    k_col_sum<<<colGrid, bl256, 0, stream>>>(P, cm);
    k_matvec_sq<<<rowGridM, bl256, 0, stream>>>(D1, rm, t1, M_, M_);
    k_matvec_sq<<<rowGridN, bl256, 0, stream>>>(D2, cm, t2, N_, N_);
    // T = D1 @ P
    k_gemm<<<gemmGrid, gemmBlock, 0, stream>>>(D1, P, T, M_, N_, M_,
        (long)M_ * M_, MN_, MN_, M_, N_, N_, 0, nullptr, nullptr, 0.f);
    // logK = -(t1 + t2 - 2*(T @ D2)) / eps   (D2 symmetric => D2^T == D2)
    k_gemm<<<gemmGrid, gemmBlock, 0, stream>>>(T, D2, LK, M_, N_, N_,
        MN_, (long)N_ * N_, MN_, N_, N_, N_, 1, t1, t2, invEps);
    // Sinkhorn (log-domain), u0 = v0 = 0
    k_fill<<<dim3((B_ * N_ + 255) / 256), bl256, 0, stream>>>(v, 0.f, B_ * N_);
    for (int s = 0; s < INNER_; ++s) {
      k_sink_row<<<rowGridM, bl256, 0, stream>>>(LK, v, u, log_a);
      k_sink_col<<<colGrid, bl256, 0, stream>>>(LK, u, v, log_b);
    }
    k_expP<<<ewGrid, bl256, 0, stream>>>(LK, u, v, P);
  }

  // ---- GW cost with final P ----
  k_row_sum<<<rowGridM, bl256, 0, stream>>>(P, rm);
  k_col_sum<<<colGrid, bl256, 0, stream>>>(P, cm);
  k_matvec_sq<<<rowGridM, bl256, 0, stream>>>(D1, rm, t1, M_, M_);
  k_matvec_sq<<<rowGridN, bl256, 0, stream>>>(D2, cm, t2, N_, N_);
  k_dot<<<dim3(B_), bl256, 0, stream>>>(rm, t1, sc + 0, M_);
  k_dot<<<dim3(B_), bl256, 0, stream>>>(cm, t2, sc + 2, N_);
  k_gemm<<<gemmGrid, gemmBlock, 0, stream>>>(D1, P, T, M_, N_, M_,
      (long)M_ * M_, MN_, MN_, M_, N_, N_, 0, nullptr, nullptr, 0.f);
  k_gemm<<<gemmGrid, gemmBlock, 0, stream>>>(P, D2, LK, M_, N_, N_,
      MN_, (long)N_ * N_, MN_, N_, N_, N_, 0, nullptr, nullptr, 0.f);
  k_cross<<<dim3(B_), bl256, 0, stream>>>(T, LK, sc + 4);
  k_final<<<dim3(1), dim3(32), 0, stream>>>(sc, out);
  k_copy<<<ewGrid, bl256, 0, stream>>>(P, out + 2, totP);
}